// Layer_7610682048830
// MI455X (gfx1250) — compile-verified
//
#include <hip/hip_runtime.h>
#include <hip/hip_bf16.h>

// MI455X (gfx1250) implementation.
// Pipeline:
//   K1 score_rows   : t[s] = <batch[s,:], pool_w>            (32 MB pass)
//   K2 softmax_s    : p[s] = softmax(t)[s]                   (tiny)
//   K3 pool_reduce  : pooled[d] = sum_s p[s]*batch[s,d]      (32 MB pass)
//   K4 gemv_wmma    : logits[v] = 0.7*(pooled @ W + b)[v]    (524 MB pass)
//        - V_WMMA_F32_16X16X4_F32 as the MAC engine
//        - W streamed via double-buffered GLOBAL_LOAD_ASYNC_TO_LDS_B128
//          synchronized with s_wait_asynccnt (ASYNCcnt split counter)
//   K5 topp_sample  : softmax over V, binary-search top-p threshold, Gumbel-max draw
// Bandwidth floor ~590MB / 23.3 TB/s ~= 25 us; W is read exactly once.

typedef __attribute__((ext_vector_type(2))) float v2f;
typedef __attribute__((ext_vector_type(8))) float v8f;
typedef int v4i_g __attribute__((vector_size(16)));  // matches builtin param type

#define TEMPERATURE 0.7f
#define TOP_P 0.5f
#define NEG_INF (-1e30f)

// ---------------------------------------------------------------- CDNA5 async helpers
__device__ __forceinline__ void async_load_b128(const float* gsrc, float* ldst) {
#if __has_builtin(__builtin_amdgcn_global_load_async_to_lds_b128)
  __builtin_amdgcn_global_load_async_to_lds_b128(
      (__attribute__((address_space(1))) v4i_g*)gsrc,
      (__attribute__((address_space(3))) v4i_g*)ldst, 0, 0);
#else
  unsigned laddr = (unsigned)(size_t)(__attribute__((address_space(3))) void*)ldst;
  asm volatile("global_load_async_to_lds_b128 %0, %1, off"
               :: "v"(laddr), "v"(gsrc)
               : "memory");
#endif
}

template <int N>
__device__ __forceinline__ void wait_async() {
#if __has_builtin(__builtin_amdgcn_s_wait_asynccnt)
  __builtin_amdgcn_s_wait_asynccnt((unsigned short)N);
#else
  asm volatile("s_wait_asynccnt %0" :: "i"(N) : "memory");
#endif
}

// ---------------------------------------------------------------- reductions
__device__ __forceinline__ float block_reduce_sum(float v, float* red) {
  for (int off = 16; off > 0; off >>= 1) v += __shfl_xor(v, off, 32);
  const int lane = threadIdx.x & 31, wv = threadIdx.x >> 5, nw = blockDim.x >> 5;
  if (lane == 0) red[wv] = v;
  __syncthreads();
  if (wv == 0) {
    float r = (lane < nw) ? red[lane] : 0.f;
    for (int off = 16; off > 0; off >>= 1) r += __shfl_xor(r, off, 32);
    if (lane == 0) red[0] = r;
  }
  __syncthreads();
  float res = red[0];
  __syncthreads();
  return res;
}

__device__ __forceinline__ float block_reduce_max(float v, float* red) {
  for (int off = 16; off > 0; off >>= 1) v = fmaxf(v, __shfl_xor(v, off, 32));
  const int lane = threadIdx.x & 31, wv = threadIdx.x >> 5, nw = blockDim.x >> 5;
  if (lane == 0) red[wv] = v;
  __syncthreads();
  if (wv == 0) {
    float r = (lane < nw) ? red[lane] : NEG_INF;
    for (int off = 16; off > 0; off >>= 1) r = fmaxf(r, __shfl_xor(r, off, 32));
    if (lane == 0) red[0] = r;
  }
  __syncthreads();
  float res = red[0];
  __syncthreads();
  return res;
}

// ---------------------------------------------------------------- K1: attention scores
__global__ void score_rows(const float* __restrict__ batch,
                           const float* __restrict__ pool_w,
                           float* __restrict__ t, int D) {
  const int row = blockIdx.x;
  const float* r = batch + (size_t)row * D;
  float acc = 0.f;
  for (int d = threadIdx.x * 4; d < D; d += blockDim.x * 4) {
    float4 a = *(const float4*)(r + d);
    float4 w = *(const float4*)(pool_w + d);
    acc += a.x * w.x + a.y * w.y + a.z * w.z + a.w * w.w;
  }
  __shared__ float red[32];
  float s = block_reduce_sum(acc, red);
  if (threadIdx.x == 0) t[row] = s;
}

// ---------------------------------------------------------------- K2: softmax over S
__global__ void softmax_s(const float* __restrict__ t, float* __restrict__ p, int S) {
  __shared__ float red[32];
  float m = NEG_INF;
  for (int s = threadIdx.x; s < S; s += blockDim.x) m = fmaxf(m, t[s]);
  m = block_reduce_max(m, red);
  float sum = 0.f;
  for (int s = threadIdx.x; s < S; s += blockDim.x) sum += __expf(t[s] - m);
  sum = block_reduce_sum(sum, red);
  const float inv = 1.f / sum;
  for (int s = threadIdx.x; s < S; s += blockDim.x) p[s] = __expf(t[s] - m) * inv;
}

// ---------------------------------------------------------------- K3: pooled vector
__global__ void pool_reduce(const float* __restrict__ batch,
                            const float* __restrict__ p,
                            float* __restrict__ pooled, int S, int D) {
  const int base = (blockIdx.x * blockDim.x + threadIdx.x) * 4;
  if (base >= D) return;
  float4 acc = make_float4(0.f, 0.f, 0.f, 0.f);
  for (int s = 0; s < S; ++s) {
    if (s + 8 < S) __builtin_prefetch(batch + (size_t)(s + 8) * D + base, 0, 0);
    const float ps = p[s];
    float4 v = *(const float4*)(batch + (size_t)s * D + base);
    acc.x += ps * v.x; acc.y += ps * v.y; acc.z += ps * v.z; acc.w += ps * v.w;
  }
  *(float4*)(pooled + base) = acc;
}

// ---------------------------------------------------------------- K4: WMMA f32 GEMV
// Block = 256 threads (8 waves), owns a 128-column strip of V.
// W is staged through a double-buffered LDS tile of 16 rows x 128 cols
// (row stride padded to 132 words) using async B128 copies, one chunk in
// flight at all times. Each wave runs 4 WMMAs per chunk on two independent
// accumulators. A(16x4) = pooled chunk broadcast to all M rows, so row M=0
// (c[0] of lanes 0..15) is the logits tile.
#define KCH 16          // K rows per chunk
#define BSTR 132        // padded LDS row stride (words)

__device__ __forceinline__ void stage_chunk(const float* __restrict__ W, int V,
                                            int v0b, int d0, float* buf, int tid) {
  const int lane = tid & 31;
  const int row = tid >> 5;  // 0..7 for 256 threads
  const float* g0 = W + (size_t)(d0 + row) * V + v0b + lane * 4;
  const float* g1 = W + (size_t)(d0 + row + 8) * V + v0b + lane * 4;
  async_load_b128(g0, buf + row * BSTR + lane * 4);
  async_load_b128(g1, buf + (row + 8) * BSTR + lane * 4);
}

__global__ __launch_bounds__(256) void gemv_wmma(const float* __restrict__ W,
                                                 const float* __restrict__ bias,
                                                 const float* __restrict__ pooled,
                                                 float* __restrict__ logits,
                                                 int D, int V) {
  __shared__ __align__(16) float sp[2048];
  __shared__ __align__(16) float kbuf[2][KCH * BSTR];

  const int tid = threadIdx.x;
  for (int i = tid; i < D && i < 2048; i += blockDim.x) sp[i] = pooled[i];

  const int v0b = blockIdx.x * 128;   // host guarantees full 128-col strips
  const int wave = tid >> 5;
  const int lane = tid & 31;
  const int v0 = v0b + wave * 16;
  const int nchunks = D / KCH;

  stage_chunk(W, V, v0b, 0, kbuf[0], tid);
  if (nchunks > 1) stage_chunk(W, V, v0b, KCH, kbuf[1], tid);

  // f32 A 16x4 layout: lanes 0-15 hold K={0,1}, lanes 16-31 hold K={2,3};
  // B 4x16 mirrors it (VGPR0 = rows K=0/2, VGPR1 = rows K=1/3).
  const int koff = (lane < 16) ? 0 : 2;
  const int n = lane & 15;
  const int bbase = wave * 16 + n;

  v8f acc0 = {};
  v8f acc1 = {};
  for (int c = 0; c < nchunks; ++c) {
    if (c + 1 < nchunks) wait_async<2>();  // chunk c landed, c+1 may be in flight
    else                 wait_async<0>();  // last chunk: drain
    __syncthreads();                       // LDS written by other threads' copies

    const float* bufc = kbuf[c & 1];
    const int d0 = c * KCH;
#pragma unroll
    for (int kk = 0; kk < KCH; kk += 8) {
      v2f a0, b0, a1, b1;
      a0.x = sp[d0 + kk + koff];
      a0.y = sp[d0 + kk + koff + 1];
      b0.x = bufc[(kk + koff) * BSTR + bbase];
      b0.y = bufc[(kk + koff + 1) * BSTR + bbase];

      a1.x = sp[d0 + kk + 4 + koff];
      a1.y = sp[d0 + kk + 4 + koff + 1];
      b1.x = bufc[(kk + 4 + koff) * BSTR + bbase];
      b1.y = bufc[(kk + 4 + koff + 1) * BSTR + bbase];

      acc0 = __builtin_amdgcn_wmma_f32_16x16x4_f32(false, a0, false, b0,
                                                   (short)0, acc0, false, false);
      acc1 = __builtin_amdgcn_wmma_f32_16x16x4_f32(false, a1, false, b1,
                                                   (short)0, acc1, false, false);
    }
    __syncthreads();                       // everyone done reading buf[c&1]
    if (c + 2 < nchunks) stage_chunk(W, V, v0b, (c + 2) * KCH, kbuf[c & 1], tid);
  }

  v8f acc = acc0 + acc1;
  if (v0 < V && lane < 16)
    logits[v0 + lane] = TEMPERATURE * (acc[0] + bias[v0 + lane]);
}

// ---------------------------------------------------------------- K5: softmax(V) + top-p + sample
__device__ __forceinline__ unsigned rotl32(unsigned x, int nb) {
  return (x << nb) | (x >> (32 - nb));
}

__device__ __forceinline__ void threefry2x32(unsigned k0, unsigned k1,
                                             unsigned c0, unsigned c1,
                                             unsigned& o0, unsigned& o1) {
  const unsigned R0[4] = {13u, 15u, 26u, 6u};
  const unsigned R1[4] = {17u, 29u, 16u, 24u};
  unsigned ks[3] = {k0, k1, k0 ^ k1 ^ 0x1BD11BDAu};
  unsigned x0 = c0 + k0, x1 = c1 + k1;
  for (int r = 0; r < 5; ++r) {
    const unsigned* rot = (r & 1) ? R1 : R0;
    for (int i = 0; i < 4; ++i) {
      x0 += x1;
      x1 = rotl32(x1, rot[i]);
      x1 ^= x0;
    }
    x0 += ks[(r + 1) % 3];
    x1 += ks[(r + 2) % 3] + (unsigned)(r + 1);
  }
  o0 = x0; o1 = x1;
}

__global__ void topp_sample(const float* __restrict__ logits,
                            float* __restrict__ probs, int V,
                            int* __restrict__ out) {
  __shared__ float red[32];
  float m = NEG_INF;
  for (int v = threadIdx.x; v < V; v += blockDim.x) m = fmaxf(m, logits[v]);
  m = block_reduce_max(m, red);
  float sum = 0.f;
  for (int v = threadIdx.x; v < V; v += blockDim.x) sum += __expf(logits[v] - m);
  sum = block_reduce_sum(sum, red);
  const float inv = 1.f / sum;
  for (int v = threadIdx.x; v < V; v += blockDim.x)
    probs[v] = __expf(logits[v] - m) * inv;
  __syncthreads();

  // binary-search threshold tau with sum_{p > tau} <= TOP_P (nucleus set)
  float lo = 0.f, hi = 1.f;
  for (int it = 0; it < 30; ++it) {
    const float mid = 0.5f * (lo + hi);
    float s = 0.f;
    for (int v = threadIdx.x; v < V; v += blockDim.x) {
      const float p = probs[v];
      if (p > mid) s += p;
    }
    s = block_reduce_sum(s, red);
    if (s > TOP_P) lo = mid; else hi = mid;  // uniform across block
  }
  const float tau = hi;

  // Gumbel-max categorical draw over the kept set (threefry2x32, key 42)
  float best = NEG_INF;
  int bidx = 0;
  for (int v = threadIdx.x; v < V; v += blockDim.x) {
    const float p = probs[v];
    if (p > tau) {
      unsigned r0, r1;
      threefry2x32(0u, 42u, (unsigned)v, 0u, r0, r1);
      float u = fmaxf((float)(r0 >> 8) * 5.9604645e-8f, 1e-12f);
      const float g = -logf(-logf(u));
      const float sc = logf(p) + g;
      if (sc > best) { best = sc; bidx = v; }
    }
  }
  for (int off = 16; off > 0; off >>= 1) {
    float ov = __shfl_xor(best, off, 32);
    int   oi = __shfl_xor(bidx, off, 32);
    if (ov > best) { best = ov; bidx = oi; }
  }
  __shared__ float rv[32];
  __shared__ int ri[32];
  const int lane = threadIdx.x & 31, wv = threadIdx.x >> 5, nw = blockDim.x >> 5;
  if (lane == 0) { rv[wv] = best; ri[wv] = bidx; }
  __syncthreads();
  if (threadIdx.x == 0) {
    float bb = rv[0]; int bi = ri[0];
    for (int i = 1; i < nw; ++i)
      if (rv[i] > bb) { bb = rv[i]; bi = ri[i]; }
    out[0] = bi;
  }
}

// ---------------------------------------------------------------- host
extern "C" void kernel_launch(void* const* d_in, const int* in_sizes, int n_in,
                              void* d_out, int out_size, void* d_ws, size_t ws_size,
                              hipStream_t stream) {
  const float* batch  = (const float*)d_in[0];  // [1,S,D]
  const float* pool_w = (const float*)d_in[1];  // [D]
  const float* W      = (const float*)d_in[2];  // [D,V]
  const float* bias   = (const float*)d_in[3];  // [V]

  const int D = in_sizes[1];
  const int S = in_sizes[0] / D;  // B == 1
  const int V = in_sizes[3];

  float* ws     = (float*)d_ws;
  float* t      = ws;             // S raw scores
  float* p      = t + S;          // S softmax weights
  float* pooled = p + S;          // D
  float* logits = pooled + D;     // V
  float* probs  = logits + V;     // V

  score_rows<<<S, 256, 0, stream>>>(batch, pool_w, t, D);
  softmax_s<<<1, 1024, 0, stream>>>(t, p, S);
  pool_reduce<<<(D + 1023) / 1024, 256, 0, stream>>>(batch, p, pooled, S, D);

  // V = 64000 = 500 full 128-column strips (reference shapes guarantee this)
  gemv_wmma<<<V / 128, 256, 0, stream>>>(W, bias, pooled, logits, D, V);

  topp_sample<<<1, 1024, 0, stream>>>(logits, probs, V, (int*)d_out);
}